// CustomConsistencyLoss_10488310137062
// MI455X (gfx1250) — compile-verified
//
#include <hip/hip_runtime.h>

// Problem constants from the reference: B=1024, H=200, W=48
#define HDIM 200
#define WDIM 48
#define HW   9600          // H*W
#define BLK  256           // 8 wave32s per block

typedef __attribute__((ext_vector_type(2))) float v2f;
typedef __attribute__((ext_vector_type(8))) float v8f;

// Full wave32 sum via V_WMMA_F32_16X16X4_F32 with B = ones.
// A layout (ISA 7.12.2, 32-bit A 16x4): lanes0-15 VGPR0 -> A[m][0], lanes16-31
// VGPR0 -> A[m][2]; VGPR1 -> K=1/K=3 (we feed 0). With B==1 everywhere:
//   D[m][n] = A[m][0]+A[m][2] = v[m] + v[m+16]   (independent of B layout)
// D layout: VGPR r holds D[r][lane] for lanes0-15 and D[8+r][lane-16] for
// lanes16-31. Summing the 8 D VGPRs gives half-sums; one xor-16 shuffle
// completes the 32-lane total (broadcast to all lanes).
__device__ __forceinline__ float wave_sum32(float v) {
    v2f a; a[0] = v;    a[1] = 0.0f;
    v2f b; b[0] = 1.0f; b[1] = 1.0f;
    v8f c = {0.f, 0.f, 0.f, 0.f, 0.f, 0.f, 0.f, 0.f};
    v8f d = __builtin_amdgcn_wmma_f32_16x16x4_f32(
        /*neg_a=*/false, a, /*neg_b=*/false, b,
        /*c_mod=*/(short)0, c, /*reuse_a=*/false, /*reuse_b=*/false);
    float s = ((d[0] + d[1]) + (d[2] + d[3])) + ((d[4] + d[5]) + (d[6] + d[7]));
    s += __shfl_xor(s, 16, 32);
    return s;
}

// One block per batch: stage hm+mask slice (2 x 38.4KB) into LDS with B128
// copies, then stream the 3 roi channels coalesced and gather from LDS.
__global__ void loss_partials_kernel(const float* __restrict__ hm,
                                     const float* __restrict__ roi,
                                     const float* __restrict__ msk,
                                     float* __restrict__ partL,
                                     float* __restrict__ partW) {
    extern __shared__ __align__(16) float smem[];
    float* s_hm = smem;        // 9600 floats
    float* s_mk = smem + HW;   // 9600 floats

    const int b   = blockIdx.x;
    const int tid = threadIdx.x;

    // ---- Phase 1: stage per-batch heightmap + mask into LDS (B128) ----
    const float4* ghm = (const float4*)(hm + (size_t)b * HW);
    const float4* gmk = (const float4*)(msk + (size_t)b * HW);
    float4* s4h = (float4*)s_hm;
    float4* s4m = (float4*)s_mk;
#pragma unroll 2
    for (int j = tid; j < HW / 4; j += BLK) {
        s4h[j] = ghm[j];
        s4m[j] = gmk[j];
    }
    __syncthreads();

    // ---- Phase 2: stream roi channels (coalesced), gather from LDS ----
    const float* ry = roi + (size_t)b * (3 * HW);
    const float* rx = ry + HW;
    const float* rv = rx + HW;

    float accL = 0.0f, accW = 0.0f;
    for (int i = tid; i < HW; i += BLK) {
        float yf = ry[i];
        float xf = rx[i];
        float vf = rv[i];
        int y = (int)yf;                     // trunc-toward-zero == reference
        int x = (int)xf;
        bool valid = (y >= 0) & (y < HDIM) & (x >= 0) & (x < WDIM);
        int yc = min(max(y, 0), HDIM - 1);
        int xc = min(max(x, 0), WDIM - 1);
        int idx = yc * WDIM + xc;
        float h = s_hm[idx];
        float m = s_mk[idx];
        float d  = h - vf;
        float ad = fabsf(d);
        float sl1 = (ad < 1.0f) ? (0.5f * d * d) : (ad - 0.5f);
        float w = valid ? m : 0.0f;
        accL = fmaf(sl1, w, accL);
        accW += w;
    }

    // ---- Phase 3: wave32 WMMA reduction, then cross-wave via LDS ----
    float wl = wave_sum32(accL);
    float ww = wave_sum32(accW);

    __shared__ float redL[BLK / 32];
    __shared__ float redW[BLK / 32];
    const int wave = tid >> 5;
    const int lane = tid & 31;
    if (lane == 0) { redL[wave] = wl; redW[wave] = ww; }
    __syncthreads();
    if (tid == 0) {
        float L = 0.0f, W = 0.0f;
#pragma unroll
        for (int k = 0; k < BLK / 32; ++k) { L += redL[k]; W += redW[k]; }
        partL[b] = L;
        partW[b] = W;
    }
}

// Deterministic final reduction over per-batch partials; writes the scalar.
__global__ void finalize_kernel(const float* __restrict__ partL,
                                const float* __restrict__ partW,
                                float* __restrict__ out, int nb) {
    const int tid = threadIdx.x;
    float accL = 0.0f, accW = 0.0f;
    for (int i = tid; i < nb; i += BLK) {
        accL += partL[i];
        accW += partW[i];
    }
    float wl = wave_sum32(accL);
    float ww = wave_sum32(accW);

    __shared__ float redL[BLK / 32];
    __shared__ float redW[BLK / 32];
    const int wave = tid >> 5;
    const int lane = tid & 31;
    if (lane == 0) { redL[wave] = wl; redW[wave] = ww; }
    __syncthreads();
    if (tid == 0) {
        float L = 0.0f, W = 0.0f;
#pragma unroll
        for (int k = 0; k < BLK / 32; ++k) { L += redL[k]; W += redW[k]; }
        out[0] = L / (W + 1e-6f);
    }
}

extern "C" void kernel_launch(void* const* d_in, const int* in_sizes, int n_in,
                              void* d_out, int out_size, void* d_ws, size_t ws_size,
                              hipStream_t stream) {
    const float* hm  = (const float*)d_in[0];  // (B,1,H,W)
    const float* roi = (const float*)d_in[1];  // (B,3,H,W)
    const float* msk = (const float*)d_in[2];  // (B,1,H,W)
    float* out = (float*)d_out;

    const int B = in_sizes[0] / HW;            // 1024

    float* partL = (float*)d_ws;               // B floats
    float* partW = partL + B;                  // B floats

    const size_t lds_bytes = (size_t)2 * HW * sizeof(float);  // 76.8 KB / WG

    hipLaunchKernelGGL(loss_partials_kernel, dim3(B), dim3(BLK), lds_bytes, stream,
                       hm, roi, msk, partL, partW);
    hipLaunchKernelGGL(finalize_kernel, dim3(1), dim3(BLK), 0, stream,
                       partL, partW, out, B);
}